// SpikeFP32Tanh_23407571764117
// MI455X (gfx1250) — compile-verified
//
#include <hip/hip_runtime.h>
#include <stdint.h>

// SNN FP32 bit-plane tanh gate for MI455X (gfx1250, wave32).
// 1 GiB total traffic -> HBM-bound (~46us floor @ 23.3 TB/s). Strategy:
//   - async global->LDS staging (ASYNCcnt path), double-buffered per wave
//   - wave32 ballot packs 32 bit-plane floats -> one u32 per op,
//     deposited per-lane with v_writelane_b32 (SGPR -> lane j, 1 VALU)
//   - exact IEEE bit reconstruction, tanh via (e^{2v}-1)/(e^{2v}+1)
//   - per-lane bit unpack + b128 coalesced stores

#define WAVES_PER_BLOCK 4
#define BLOCK_THREADS   (WAVES_PER_BLOCK * 32)
#define GROUP_VALS      32
#define GROUP_FLOATS    (GROUP_VALS * 32)      // 1024 floats
#define GROUP_BYTES     (GROUP_FLOATS * 4)     // 4 KB per group

#define LDS_ASPACE __attribute__((address_space(3)))

// Exact LDS byte offset of a __shared__ object, via generic->AS(3) cast.
// The ptrtoint also CAPTURES the pointer, so the "memory"-clobbering inline
// asm below is treated as a potential writer of smem -> ds loads survive.
__device__ __forceinline__ unsigned lds_byte_offset(const void* p) {
    return (unsigned)(uintptr_t)(LDS_ASPACE const void*)p;
}

__global__ __launch_bounds__(BLOCK_THREADS)
void snn_tanh_bits_kernel(const float* __restrict__ x,
                          float* __restrict__ y,
                          int n_groups, int groups_per_wave)
{
    // 2 buffers x 4 waves x 4KB = 32 KB static LDS
    __shared__ float smem[2 * WAVES_PER_BLOCK * GROUP_FLOATS];

    const int lane = threadIdx.x & 31;
    const int wid  = threadIdx.x >> 5;
    const long long wave_id = (long long)blockIdx.x * WAVES_PER_BLOCK + wid;

    long long gBegin = wave_id * (long long)groups_per_wave;
    long long gEnd   = gBegin + groups_per_wave;
    if (gEnd > (long long)n_groups) gEnd = n_groups;
    if (gBegin >= gEnd) return;   // wave-uniform exit

    // ---- async stage: one 4KB group -> this wave's LDS slice ---------------
    auto stage = [&](long long g, int buf) {
        const float* lp = &smem[buf * (WAVES_PER_BLOCK * GROUP_FLOATS)
                                + wid * GROUP_FLOATS + lane * 4];
        unsigned lds  = lds_byte_offset(lp);                      // per-lane 16B slot
        unsigned goff = (unsigned)(g * GROUP_BYTES + lane * 16);  // < 2^29, fits i32
#pragma unroll
        for (int k = 0; k < 8; ++k) {
            // GVS mode: saddr(base) + vaddr(32-bit offset) + inst offset.
            // INST_OFFSET advances both the global and the LDS address.
            asm volatile("global_load_async_to_lds_b128 %0, %1, %2 offset:%3"
                         :: "v"(lds), "v"(goff), "s"(x), "i"(k * 512)
                         : "memory");
        }
    };

    stage(gBegin, 0);  // prime the pipeline (8 async ops in flight)

    for (long long g = gBegin; g < gEnd; ++g) {
        const int buf = (int)((g - gBegin) & 1);

        if (g + 1 < gEnd) {
            stage(g + 1, buf ^ 1);                                // +8 in flight
            asm volatile("s_wait_asynccnt 0x8" ::: "memory");     // current group done
        } else {
            asm volatile("s_wait_asynccnt 0x0" ::: "memory");     // drain
        }

        // ---- pack: 32 ballots -> each lane owns value (g*32 + lane) --------
        const float* sm = smem + buf * (WAVES_PER_BLOCK * GROUP_FLOATS)
                               + wid * GROUP_FLOATS;
        int myu = 0;
#pragma unroll
        for (int j = 0; j < 32; ++j) {
            float f = sm[j * 32 + lane];                 // ds_load (2addr-paired)
            unsigned m = __builtin_amdgcn_ballot_w32(f != 0.0f);
            // SGPR ballot -> lane j of accumulator, single v_writelane_b32
            asm volatile("v_writelane_b32 %0, %1, %2"
                         : "+v"(myu) : "s"(m), "i"(j));
        }
        unsigned u = __builtin_bitreverse32((unsigned)myu); // MSB-first -> IEEE bits

        // ---- tanh (reference formula) --------------------------------------
        float v   = __uint_as_float(u);
        float e2x = expf(2.0f * v);
        float t   = (e2x - 1.0f) / (e2x + 1.0f);

        // ---- encode (reference maps zero/subnormal to sign|0x00800000) -----
        unsigned ou = __float_as_uint(t);
        if ((ou & 0x7f800000u) == 0u)
            ou = ((t < 0.0f) ? 0x80000000u : 0u) | 0x00800000u;

        // ---- unpack: this lane writes its value's 32 output bits -----------
        float4* dst = (float4*)(y + (g * GROUP_VALS + lane) * 32);
#pragma unroll
        for (int k = 0; k < 8; ++k) {
            float4 o;
            o.x = (float)((ou >> (31 - (4 * k + 0))) & 1u);
            o.y = (float)((ou >> (31 - (4 * k + 1))) & 1u);
            o.z = (float)((ou >> (31 - (4 * k + 2))) & 1u);
            o.w = (float)((ou >> (31 - (4 * k + 3))) & 1u);
            dst[k] = o;                                   // global_store_b128
        }
    }
}

extern "C" void kernel_launch(void* const* d_in, const int* in_sizes, int n_in,
                              void* d_out, int out_size, void* d_ws, size_t ws_size,
                              hipStream_t stream) {
    (void)n_in; (void)out_size; (void)d_ws; (void)ws_size;
    const float* x = (const float*)d_in[0];
    float*       y = (float*)d_out;

    const long long n_bits = (long long)in_sizes[0];     // 4096*1024*32
    const long long n_vals = n_bits / 32;
    const int n_groups = (int)(n_vals / GROUP_VALS);     // 131072 groups of 32 values

    // ~8 groups per wave for async double-buffering depth
    int target_waves = (n_groups + 7) / 8;
    int blocks = (target_waves + WAVES_PER_BLOCK - 1) / WAVES_PER_BLOCK;   // 4096
    if (blocks < 1) blocks = 1;
    int total_waves = blocks * WAVES_PER_BLOCK;
    int gpw = (n_groups + total_waves - 1) / total_waves;                  // 8
    if (gpw < 1) gpw = 1;

    snn_tanh_bits_kernel<<<dim3(blocks), dim3(BLOCK_THREADS), 0, stream>>>(
        x, y, n_groups, gpw);
}